// UpBlock_66073776881957
// MI455X (gfx1250) — compile-verified
//
#include <hip/hip_runtime.h>

typedef __attribute__((ext_vector_type(2))) float v2f;
typedef __attribute__((ext_vector_type(8))) float v8f;

#define CIN   64
#define KK    27
#define EPSBN 1e-5f

// ---------------------------------------------------------------------------
// Zero a small float buffer (used for the shared "missing voxel" zero row).
// ---------------------------------------------------------------------------
__global__ __launch_bounds__(64) void zero_buf(float* __restrict__ p, int n) {
  int i = blockIdx.x * blockDim.x + threadIdx.x;
  if (i < n) p[i] = 0.f;
}

// ---------------------------------------------------------------------------
// Repack W[k][cin][ncols] -> Wp[(k*(CIN/2) + cin/2)*ncols + n] = {W[k][cin][n],
// W[k][cin+1][n]} so the WMMA B operand is a single 8-byte load per lane.
// ---------------------------------------------------------------------------
__global__ __launch_bounds__(256) void repack_w(const float* __restrict__ W,
                                                v2f* __restrict__ Wp,
                                                int ncols) {
  int i = blockIdx.x * blockDim.x + threadIdx.x;
  int total = KK * (CIN / 2) * ncols;
  if (i >= total) return;
  int n  = i % ncols;
  int t  = i / ncols;
  int c2 = t % (CIN / 2);
  int k  = t / (CIN / 2);
  const float* Wk = W + (size_t)k * CIN * ncols;
  v2f p;
  p.x = Wk[(2 * c2 + 0) * ncols + n];
  p.y = Wk[(2 * c2 + 1) * ncols + n];
  Wp[i] = p;
}

// ---------------------------------------------------------------------------
// Gather-GEMM-scatter sparse conv + BN + ReLU (+ optional residual add).
// One wave32 computes an (MT*16) x NCOLS output block with
// V_WMMA_F32_16X16X4_F32. Register tiling:
//   - the A fragment of each row-tile is reused across all NCOLS/16 col-tiles
//   - the B fragment is reused across all MT row-tiles
// => per K=4 step: MT A-loads + TN B-loads feed MT*TN WMMAs
//    (0.75 loads/WMMA for NCOLS=64, MT=2).
//   feat : [nrows_in, CIN] gather source; index == nrows_in -> zero row
//   zrow : CIN floats of zeros (pointer-selected per k, no per-load cndmask)
//   nmap : [KK, P] int32 ; Wp repacked weights ; out : [P, NCOLS]
//   P and NCOLS are multiples of 16. Out-of-range row-tiles are clamped to
//   the last tile (duplicate identical work/stores; benign).
// ---------------------------------------------------------------------------
template <int NCOLS, int MT, bool ADD_PREV>
__global__ __launch_bounds__(256) void sparse_conv_wmma(
    const float* __restrict__ feat, int nrows_in,
    const float* __restrict__ zrow,
    const int* __restrict__ nmap, int P,
    const v2f* __restrict__ Wp,
    const float* __restrict__ bias,
    const float* __restrict__ gamma, const float* __restrict__ beta,
    const float* __restrict__ mean,  const float* __restrict__ var,
    const float* __restrict__ prev,
    float* __restrict__ out) {
  constexpr int TN = NCOLS / 16;  // column tiles per wave

  const int wave    = (blockIdx.x * blockDim.x + threadIdx.x) >> 5;
  const int tiles_m = P >> 4;
  const int nwaves  = (tiles_m + MT - 1) / MT;
  if (wave >= nwaves) return;  // wave-uniform guard -> s_cbranch_execz

  const int lane = threadIdx.x & 31;
  const int half = lane >> 4;  // 0 -> K slots {0,1}, 1 -> {2,3}
  const int lm   = lane & 15;

  int tile_m[MT];
#pragma unroll
  for (int mt = 0; mt < MT; ++mt) {
    int t = wave * MT + mt;
    tile_m[mt] = t < tiles_m ? t : tiles_m - 1;  // clamp tail (duplicate work)
  }

  v8f acc[MT][TN];
#pragma unroll
  for (int mt = 0; mt < MT; ++mt)
#pragma unroll
    for (int t = 0; t < TN; ++t) acc[mt][t] = (v8f){};

  for (int k = 0; k < KK; ++k) {
    const float* arow[MT];
#pragma unroll
    for (int mt = 0; mt < MT; ++mt) {
      const int ridx = nmap[(size_t)k * P + tile_m[mt] * 16 + lm];
      // One pointer select per (k, mt) instead of per-load value selects.
      arow[mt] = ((unsigned)ridx < (unsigned)nrows_in)
                     ? feat + (size_t)ridx * CIN + 2 * half
                     : zrow + 2 * half;
    }
    const v2f* wrow = Wp + ((size_t)k * (CIN / 2) + half) * NCOLS + lm;
#pragma unroll
    for (int c = 0; c < CIN; c += 4) {  // K=4 per WMMA
      v2f a[MT];
#pragma unroll
      for (int mt = 0; mt < MT; ++mt)
        a[mt] = *(const v2f*)(arow[mt] + c);  // A[M=lm][c+2h .. c+2h+1]
#pragma unroll
      for (int t = 0; t < TN; ++t) {
        v2f b = wrow[(c / 2) * NCOLS + 16 * t];  // B[c+2h..+1][16t+lm]
#pragma unroll
        for (int mt = 0; mt < MT; ++mt)
          acc[mt][t] = __builtin_amdgcn_wmma_f32_16x16x4_f32(
              false, a[mt], false, b, (short)0, acc[mt][t], false, false);
      }
    }
  }

  // Fused epilogue: bias + BN + ReLU (+ residual). C/D layout:
  // VGPR r, lanes 0-15 -> row r, lanes 16-31 -> row r+8; column = lm.
#pragma unroll
  for (int t = 0; t < TN; ++t) {
    const int n     = 16 * t + lm;
    const float sc  = gamma[n] * rsqrtf(var[n] + EPSBN);
    const float bb  = bias[n];
    const float mm  = mean[n];
    const float bbe = beta[n];
#pragma unroll
    for (int mt = 0; mt < MT; ++mt) {
#pragma unroll
      for (int r = 0; r < 8; ++r) {
        float x = acc[mt][t][r] + bb;
        x = (x - mm) * sc + bbe;
        x = fmaxf(x, 0.f);
        const size_t oidx =
            (size_t)(tile_m[mt] * 16 + r + 8 * half) * NCOLS + n;
        if (ADD_PREV) x += prev[oidx];
        out[oidx] = x;
      }
    }
  }
}

// ---------------------------------------------------------------------------
extern "C" void kernel_launch(void* const* d_in, const int* in_sizes, int n_in,
                              void* d_out, int out_size, void* d_ws,
                              size_t ws_size, hipStream_t stream) {
  (void)n_in; (void)out_size; (void)ws_size;

  const float* feat      = (const float*)d_in[0];
  const float* skip      = (const float*)d_in[1];
  const int*   nmap_subm = (const int*)d_in[2];
  const int*   nmap_up   = (const int*)d_in[3];
  const float* W1 = (const float*)d_in[4];
  const float* b1 = (const float*)d_in[5];
  const float* g1 = (const float*)d_in[6];
  const float* be1 = (const float*)d_in[7];
  const float* m1 = (const float*)d_in[8];
  const float* v1 = (const float*)d_in[9];
  const float* W2 = (const float*)d_in[10];
  const float* b2 = (const float*)d_in[11];
  const float* g2 = (const float*)d_in[12];
  const float* be2 = (const float*)d_in[13];
  const float* m2 = (const float*)d_in[14];
  const float* v2 = (const float*)d_in[15];
  const float* Wu = (const float*)d_in[16];
  const float* bu = (const float*)d_in[17];
  const float* gu = (const float*)d_in[18];
  const float* beu = (const float*)d_in[19];
  const float* mu = (const float*)d_in[20];
  const float* vu = (const float*)d_in[21];

  const int N = in_sizes[0] / CIN;  // 150000
  const int M = in_sizes[1] / 32;   // 300000

  float* ws    = (float*)d_ws;
  float* x_net = ws;                           // [N, 64]
  float* x2    = ws + (size_t)N * CIN;         // [N, 64]
  v2f*   W1p   = (v2f*)(ws + 2 * (size_t)N * CIN);
  v2f*   W2p   = W1p + (size_t)KK * (CIN / 2) * 64;
  v2f*   Wup   = W2p + (size_t)KK * (CIN / 2) * 64;
  float* zrow  = (float*)(Wup + (size_t)KK * (CIN / 2) * 32);  // [CIN] zeros

  // Tiny prep kernels (stream-ordered): zero row + weight repack
  zero_buf<<<1, 64, 0, stream>>>(zrow, CIN);
  {
    int t64 = KK * (CIN / 2) * 64;
    int t32 = KK * (CIN / 2) * 32;
    repack_w<<<(t64 + 255) / 256, 256, 0, stream>>>(W1, W1p, 64);
    repack_w<<<(t64 + 255) / 256, 256, 0, stream>>>(W2, W2p, 64);
    repack_w<<<(t32 + 255) / 256, 256, 0, stream>>>(Wu, Wup, 32);
  }

  constexpr int MT = 2;
  // Pass 1: x_net = relu(bn1(conv(feat, nmap_subm, W1) + b1))
  {
    int waves  = (N / 16 + MT - 1) / MT;
    int blocks = (waves * 32 + 255) / 256;
    sparse_conv_wmma<64, MT, false><<<blocks, 256, 0, stream>>>(
        feat, N, zrow, nmap_subm, N, W1p, b1, g1, be1, m1, v1, nullptr, x_net);
  }
  // Pass 2: x2 = x_net + relu(bn2(conv(x_net, nmap_subm, W2) + b2))
  {
    int waves  = (N / 16 + MT - 1) / MT;
    int blocks = (waves * 32 + 255) / 256;
    sparse_conv_wmma<64, MT, true><<<blocks, 256, 0, stream>>>(
        x_net, N, zrow, nmap_subm, N, W2p, b2, g2, be2, m2, v2, x_net, x2);
  }
  // Pass 3: out = relu(bnu(conv(x2, nmap_up, Wu) + bu)) + skip
  {
    int waves  = (M / 16 + MT - 1) / MT;
    int blocks = (waves * 32 + 255) / 256;
    sparse_conv_wmma<32, MT, true><<<blocks, 256, 0, stream>>>(
        x2, N, zrow, nmap_up, M, Wup, bu, gu, beu, mu, vu, skip,
        (float*)d_out);
  }
}